// ReadHead_77240691851639
// MI455X (gfx1250) — compile-verified
//
#include <hip/hip_runtime.h>
#include <hip/hip_bf16.h>
#include <math.h>

#define EPSV 1e-8f
#define BB 64
#define NN 8192
#define WW 64
#define HH 4
#define CTRLN 280

typedef __attribute__((ext_vector_type(2))) float v2f;
typedef __attribute__((ext_vector_type(4))) float v4f;
typedef __attribute__((ext_vector_type(8))) float v8f;

__device__ __forceinline__ float softplusf(float x) {
    return (x > 0.f) ? (x + log1pf(expf(-x))) : log1pf(expf(x));
}

// ---------------- K1: control preprocessing (one thread per (b,h)) ----------
__global__ void k_ctrl(const float* __restrict__ controls,
                       float* __restrict__ keys_t, float* __restrict__ knorm,
                       float* __restrict__ betaA, float* __restrict__ gateA,
                       float* __restrict__ shiftA, float* __restrict__ gammaA) {
    int t = threadIdx.x;            // 0..255 == b*H + h
    int b = t >> 2, h = t & 3;
    const float* c = controls + (size_t)b * CTRLN;
    float* kt = keys_t + (size_t)t * WW;
    float kn = 0.f;
    for (int w = 0; w < WW; ++w) {
        float kv = tanhf(c[h * WW + w]);
        kt[w] = kv;
        kn = fmaf(kv, kv, kn);
    }
    knorm[t] = sqrtf(kn);
    betaA[t] = softplusf(c[HH * WW + h]);
    gateA[t] = 1.f / (1.f + expf(-c[HH * WW + HH + h]));
    float x0 = c[HH * WW + 2 * HH + h * 3 + 0];
    float x1 = c[HH * WW + 2 * HH + h * 3 + 1];
    float x2 = c[HH * WW + 2 * HH + h * 3 + 2];
    float mx = fmaxf(x0, fmaxf(x1, x2));
    float e0 = expf(x0 - mx), e1 = expf(x1 - mx), e2 = expf(x2 - mx);
    float inv = 1.f / (e0 + e1 + e2);
    shiftA[t * 3 + 0] = e0 * inv;
    shiftA[t * 3 + 1] = e1 * inv;
    shiftA[t * 3 + 2] = e2 * inv;
    gammaA[t] = 1.f + softplusf(c[HH * WW + 5 * HH + h]);
}

// ---------------- K2: content addressing via WMMA f32 16x16x4 ---------------
// grid (8, B), block 256 (8 waves). Each wave: 128 n-rows (8 chunks of 16).
// K-permutation: WMMA op kc consumes logical columns
//   col = (lane>>4)*32 + 2*kc + v   (v = VGPR index 0/1, hwK = (lane>>4)*2+v)
// so each lane's A/B fragments are a CONTIGUOUS 32-float run -> b128 loads.
// Every column 0..63 is covered exactly once across the 16 ops, and A/B use
// the same permutation, so the accumulated dot product is exact.
__global__ void k_content(const float* __restrict__ mem,
                          const float* __restrict__ keys_t,
                          const float* __restrict__ knorm,
                          const float* __restrict__ betaA,
                          float* __restrict__ eBuf,
                          float* __restrict__ sumE_part) {
    const int b    = blockIdx.y;
    const int tid  = threadIdx.x;
    const int lane = tid & 31;
    const int wave = tid >> 5;
    const int row  = lane & 15;          // A: n-row within chunk; C: head column
    const int kbase = (lane >> 4) * 32;  // contiguous half-row per lane half
    const int mofs = (lane >> 4) * 8;    // C: M offset
    const int nwave = blockIdx.x * 1024 + wave * 128;

    const int hh = row;
    float kb = 0.f, kn = 0.f;
    if (hh < HH) { kb = betaA[b * HH + hh]; kn = knorm[b * HH + hh]; }

    // Preload key B-fragments once: contiguous 32 floats -> 8 x b128 loads.
    // Mask-multiply instead of exec branch (lanes hh>=H read head 0, zeroed).
    const float kmsk = (hh < HH) ? 1.f : 0.f;
    const float* krow = keys_t + ((size_t)(b * HH) + (hh < HH ? hh : 0)) * WW + kbase;
    v4f kq[8];
#pragma unroll
    for (int q = 0; q < 8; ++q) {
        kq[q] = ((const v4f*)krow)[q];
        kq[q] *= kmsk;
    }

    float lsum = 0.f;
    for (int ch = 0; ch < 8; ++ch) {
        const int nb = nwave + ch * 16;
        const float* mrow = mem + ((size_t)b * NN + nb + row) * WW + kbase;
        // pull next chunk's rows toward L2 while this chunk computes
        __builtin_prefetch(mrow + 16 * WW, 0, 0);

        // Stage the lane's contiguous half-row: 8 independent b128 loads.
        v4f a4[8];
#pragma unroll
        for (int q = 0; q < 8; ++q) a4[q] = ((const v4f*)mrow)[q];

        v8f acc = {0.f, 0.f, 0.f, 0.f, 0.f, 0.f, 0.f, 0.f};
        float msq = 0.f;
#pragma unroll
        for (int q = 0; q < 8; ++q) {
            msq = fmaf(a4[q].x, a4[q].x, msq);
            msq = fmaf(a4[q].y, a4[q].y, msq);
            msq = fmaf(a4[q].z, a4[q].z, msq);
            msq = fmaf(a4[q].w, a4[q].w, msq);
            v2f a, kf;
            a.x = a4[q].x;  a.y = a4[q].y;
            kf.x = kq[q].x; kf.y = kq[q].y;
            acc = __builtin_amdgcn_wmma_f32_16x16x4_f32(
                false, a, false, kf, (short)0, acc, false, false);
            a.x = a4[q].z;  a.y = a4[q].w;
            kf.x = kq[q].z; kf.y = kq[q].w;
            acc = __builtin_amdgcn_wmma_f32_16x16x4_f32(
                false, a, false, kf, (short)0, acc, false, false);
        }
        // full row normsq: lane L holds half of row (L&15); fold with lane^16
        float full = msq + __shfl_xor(msq, 16, 32);
#pragma unroll
        for (int r = 0; r < 8; ++r) {
            float mn2 = __shfl(full, mofs + r, 32);  // ||mem[nb+mofs+r]||^2
            float score = acc[r] / (kn * sqrtf(mn2) + EPSV) * kb;
            float ev = expf(score);   // bounded: |cos|*beta small, no max-sub
            if (hh < HH) {
                eBuf[((size_t)(b * HH + hh)) * NN + (nb + mofs + r)] = ev;
                lsum += ev;
            }
        }
    }
    lsum += __shfl_xor(lsum, 16, 32);
    if (lane < HH)
        sumE_part[(b * HH + lane) * 64 + (blockIdx.x * 8 + wave)] = lsum;
}

// ---------------- generic deterministic partial-sum reduction ---------------
__global__ void k_reduce(const float* __restrict__ part,
                         float* __restrict__ dst, int cnt) {
    int t = threadIdx.x;   // 0..255 == b*H+h
    float s = 0.f;
    for (int i = 0; i < cnt; ++i) s += part[(size_t)t * cnt + i];
    dst[t] = s;
}

// ------- K3: interpolate + circular conv + sharpen; block = 256 n of one (b,h) -------
__global__ void k_shift(const float* __restrict__ eBuf,
                        const float* __restrict__ prev,
                        const float* __restrict__ sumE,
                        const float* __restrict__ gateA,
                        const float* __restrict__ shiftA,
                        const float* __restrict__ gammaA,
                        float* __restrict__ wBuf,
                        float* __restrict__ sumS_part) {
    const int tid = threadIdx.x;
    const size_t idx = (size_t)blockIdx.x * 256 + tid;
    const int n  = (int)(idx & (NN - 1));
    const int bh = (int)(idx >> 13);
    const float* e = eBuf + (size_t)bh * NN;
    const float* p = prev + (size_t)bh * NN;
    const float g = gateA[bh], gm = gammaA[bh];
    const float s0 = shiftA[bh * 3 + 0];
    const float s1 = shiftA[bh * 3 + 1];
    const float s2 = shiftA[bh * 3 + 2];
    const float einv = g / sumE[bh];
    const float pg = 1.f - g;
    const int nm = (n == 0) ? NN - 1 : n - 1;
    const int np = (n == NN - 1) ? 0 : n + 1;
    float wm = einv * e[nm] + pg * p[nm];
    float wc = einv * e[n]  + pg * p[n];
    float wp = einv * e[np] + pg * p[np];
    float wsh = s0 * wm + s1 * wc + s2 * wp;   // out[i]=s0*w[i-1]+s1*w[i]+s2*w[i+1]
    float wpw = powf(wsh, gm);                 // wsh > 0 always
    wBuf[(size_t)bh * NN + n] = wpw;

    float v = wpw;
    for (int o = 16; o; o >>= 1) v += __shfl_xor(v, o, 32);
    __shared__ float wsum[8];
    if ((tid & 31) == 0) wsum[tid >> 5] = v;
    __syncthreads();
    if (tid == 0) {
        float s = 0.f;
        for (int i = 0; i < 8; ++i) s += wsum[i];
        sumS_part[bh * 32 + (blockIdx.x & 31)] = s;
    }
}

// ---------------- K4: read einsum via WMMA f32 16x16x4 ----------------------
// M=heads (4 of 16 used), N=w (4 tiles of 16), K=n. grid (8,B), 8 waves.
__global__ void k_read(const float* __restrict__ mem,
                       const float* __restrict__ wBuf,
                       float* __restrict__ out_part) {
    const int b    = blockIdx.y;
    const int tid  = threadIdx.x;
    const int lane = tid & 31;
    const int wave = tid >> 5;
    const int row  = lane & 15;          // A: head row; B/C: w column
    const int kg   = (lane >> 4) * 2;
    const int n0   = blockIdx.x * 1024 + wave * 128;

    // mask-multiply instead of exec branch for the A fragments
    const float amsk = (row < HH) ? 1.f : 0.f;
    const float* wrow = wBuf + ((size_t)(b * HH) + (row < HH ? row : 0)) * NN;

    v8f c0 = {0.f,0.f,0.f,0.f,0.f,0.f,0.f,0.f};
    v8f c1 = c0, c2 = c0, c3 = c0;
    for (int kc = 0; kc < 32; ++kc) {
        const int n = n0 + kc * 4 + kg;
        v2f a;
        a.x = wrow[n] * amsk;
        a.y = wrow[n + 1] * amsk;
        const float* m0 = mem + ((size_t)b * NN + n) * WW + row;
        v2f bf;
        bf.x = m0[0];       bf.y = m0[WW];
        c0 = __builtin_amdgcn_wmma_f32_16x16x4_f32(false, a, false, bf, (short)0, c0, false, false);
        bf.x = m0[16];      bf.y = m0[WW + 16];
        c1 = __builtin_amdgcn_wmma_f32_16x16x4_f32(false, a, false, bf, (short)0, c1, false, false);
        bf.x = m0[32];      bf.y = m0[WW + 32];
        c2 = __builtin_amdgcn_wmma_f32_16x16x4_f32(false, a, false, bf, (short)0, c2, false, false);
        bf.x = m0[48];      bf.y = m0[WW + 48];
        c3 = __builtin_amdgcn_wmma_f32_16x16x4_f32(false, a, false, bf, (short)0, c3, false, false);
    }
    // deterministic cross-wave reduction in LDS (M=r valid for r<4, lanes<16)
    __shared__ float part[8][HH][WW];
    if (lane < 16) {
#pragma unroll
        for (int r = 0; r < HH; ++r) {
            part[wave][r][ 0 + row] = c0[r];
            part[wave][r][16 + row] = c1[r];
            part[wave][r][32 + row] = c2[r];
            part[wave][r][48 + row] = c3[r];
        }
    }
    __syncthreads();
    float s = 0.f;
    for (int i = 0; i < 8; ++i) s += part[i][tid >> 6][tid & 63];
    out_part[((size_t)(b * 8 + blockIdx.x)) * (HH * WW) + tid] = s;
}

// ---------------- K5: combine n-tile partials + sharpen renormalize ---------
__global__ void k_final(const float* __restrict__ out_part,
                        const float* __restrict__ sumS,
                        float* __restrict__ out) {
    int t = blockIdx.x * 256 + threadIdx.x;  // 0..16383 = b*256 + h*64 + w
    int b = t >> 8;
    int hw = t & 255;
    int h = hw >> 6;
    float s = 0.f;
    for (int i = 0; i < 8; ++i) s += out_part[((size_t)(b * 8 + i)) * 256 + hw];
    out[t] = s / (sumS[b * HH + h] + EPSV);
}

extern "C" void kernel_launch(void* const* d_in, const int* in_sizes, int n_in,
                              void* d_out, int out_size, void* d_ws, size_t ws_size,
                              hipStream_t stream) {
    const float* memory   = (const float*)d_in[0];   // [B,N,W]
    const float* controls = (const float*)d_in[1];   // [B,280]
    const float* prevw    = (const float*)d_in[2];   // [B,H,N]
    float* out = (float*)d_out;                      // [B,H,W]

    float* p = (float*)d_ws;
    float* keys_t = p; p += (size_t)BB * HH * WW;    // 16384
    float* knorm  = p; p += BB * HH;                 // 256
    float* betaA  = p; p += BB * HH;
    float* gateA  = p; p += BB * HH;
    float* shiftA = p; p += BB * HH * 3;
    float* gammaA = p; p += BB * HH;
    float* sumE   = p; p += BB * HH;
    float* sumS   = p; p += BB * HH;
    float* eBuf   = p; p += (size_t)BB * HH * NN;    // 2M floats
    float* wBuf   = p; p += (size_t)BB * HH * NN;    // 2M floats
    float* sumEp  = p; p += BB * HH * 64;
    float* sumSp  = p; p += BB * HH * 32;
    float* outp   = p; p += (size_t)BB * 8 * HH * WW;

    k_ctrl<<<1, 256, 0, stream>>>(controls, keys_t, knorm, betaA, gateA, shiftA, gammaA);
    k_content<<<dim3(8, BB), 256, 0, stream>>>(memory, keys_t, knorm, betaA, eBuf, sumEp);
    k_reduce<<<1, 256, 0, stream>>>(sumEp, sumE, 64);
    k_shift<<<dim3((BB * HH * NN) / 256), 256, 0, stream>>>(eBuf, prevw, sumE, gateA,
                                                            shiftA, gammaA, wBuf, sumSp);
    k_reduce<<<1, 256, 0, stream>>>(sumSp, sumS, 32);
    k_read<<<dim3(8, BB), 256, 0, stream>>>(memory, wBuf, outp);
    k_final<<<dim3(BB), 256, 0, stream>>>(outp, sumS, out);
}